// LCAOConv_43774306681431
// MI455X (gfx1250) — compile-verified
//
#include <hip/hip_runtime.h>

// Problem constants (from the reference)
#define NN 50000
#define EE 800000
#define HH 128
#define DD 32
#define CC 64
#define RR 8

typedef float v2f __attribute__((ext_vector_type(2)));
typedef float v8f __attribute__((ext_vector_type(8)));

// LDS row strides (in floats), padded for bank-conflict avoidance.
#define STRA 132   // A/mid tiles: 16 lanes step by row -> 132%64=4 spreads 16 banks
#define STRW 136   // weight tiles: 2*136%64=16 -> the two K-halves use disjoint bank groups
#define STRWB 40   // 32-wide weight tile: 2*40%64=16 -> disjoint halves
#define STRG 36    // 32-wide A tile: 36*m mod 64 distinct for m=0..15

__device__ __forceinline__ float silu_f(float x) {
    return x / (1.0f + __expf(-x));
}

// CDNA5 async global->LDS copy of 16 bytes (GLOBAL_LOAD_ASYNC_TO_LDS_B128,
// tracked by ASYNCcnt; per-lane LDS byte address in VDST, 64-bit addr in VADDR).
__device__ __forceinline__ void async_copy_f4(const float* lds_dst, const float* gsrc) {
    uint32_t lds_addr = (uint32_t)(uintptr_t)lds_dst;   // low 32 bits = LDS offset
    uint64_t gaddr    = (uint64_t)(uintptr_t)gsrc;
    asm volatile("global_load_async_to_lds_b128 %0, %1, off"
                 :: "v"(lds_addr), "v"(gaddr) : "memory");
}
__device__ __forceinline__ void wait_async0() {
    asm volatile("s_wait_asynccnt 0x0" ::: "memory");
}

// ---------------------------------------------------------------------------
// Fused 2-layer MLP:  out[M,32] = silu( silu(in[M,K1]) @ Wa[K1,128] + ba ) @ Wb[128,32] + bb
// Block: 256 threads (8 wave32), 32 rows per block.
// Weights staged into LDS once per block via async global->LDS DMA; all
// matrix math is V_WMMA_F32_16X16X4_F32 (fp32 in / fp32 acc).
// ---------------------------------------------------------------------------
__global__ __launch_bounds__(256)
void mlp2_kernel(const float* __restrict__ in, int M, int K1,
                 const float* __restrict__ Wa, const float* __restrict__ ba,
                 const float* __restrict__ Wb, const float* __restrict__ bb,
                 float* __restrict__ out) {
    __shared__ float Ain[32 * STRA];     // silu(input) tile [32 x K1]
    __shared__ float Bmid[32 * STRA];    // silu(layer1) tile [32 x 128]
    __shared__ float WaL[HH * STRW];     // staged Wa (up to 128x128)
    __shared__ float WbL[HH * STRWB];    // staged Wb (128x32)

    const int tid  = threadIdx.x;
    const int lane = tid & 31;
    const int wave = tid >> 5;
    const int row0 = blockIdx.x * 32;

    // --- Async-stage weights (raw copy, 16B per lane per issue) -----------
    {
        const int nva = (K1 * HH) >> 2;          // vec4 count, multiple of 256
        for (int v4 = tid; v4 < nva; v4 += 256) {
            int r = v4 >> 5;                     // 128 floats = 32 vec4 per row
            int cpos = (v4 & 31) << 2;
            async_copy_f4(&WaL[r * STRW + cpos], Wa + r * HH + cpos);
        }
        const int nvb = (HH * DD) >> 2;          // 1024 vec4
        for (int v4 = tid; v4 < nvb; v4 += 256) {
            int r = v4 >> 3;                     // 32 floats = 8 vec4 per row
            int cpos = (v4 & 7) << 2;
            async_copy_f4(&WbL[r * STRWB + cpos], Wb + r * DD + cpos);
        }
    }

    // --- Stage silu(in) tile (needs VALU transform -> classic path) -------
    for (int idx = tid; idx < 32 * K1; idx += 256) {
        int r = idx / K1, cc = idx - r * K1;
        int grow = row0 + r;
        float v = (grow < M) ? in[(long)grow * K1 + cc] : 0.0f;
        Ain[r * STRA + cc] = silu_f(v);
    }
    wait_async0();        // our async copies landed in LDS
    __syncthreads();      // everyone's staging visible

    const int m  = lane & 15;   // row (A) / col (B,D) index within 16-tile
    const int kh = lane >> 4;   // K-half selector per ISA fragment layout

    // ---- GEMM1: [32,K1] @ [K1,128]; wave w owns column tile w, both row tiles
    const int ct   = wave;
    const int colg = ct * 16 + m;
    const float bias1 = ba ? ba[colg] : 0.0f;
    for (int rt = 0; rt < 2; ++rt) {
        v8f acc = {};
        for (int k0 = 0; k0 < K1; k0 += 4) {
            v2f a, b;
            // A 16x4 fp32: lanes 0-15 -> K=k0,k0+1 ; lanes 16-31 -> K=k0+2,k0+3
            a.x = Ain[(rt * 16 + m) * STRA + k0 + 2 * kh];
            a.y = Ain[(rt * 16 + m) * STRA + k0 + 2 * kh + 1];
            // B 4x16 fp32 from LDS-staged weights
            b.x = WaL[(k0 + 2 * kh) * STRW + colg];
            b.y = WaL[(k0 + 2 * kh + 1) * STRW + colg];
            acc = __builtin_amdgcn_wmma_f32_16x16x4_f32(
                false, a, false, b, (short)0, acc, false, false);
        }
        // D 16x16 fp32: VGPR r -> M=r (lanes 0-15), M=r+8 (lanes 16-31)
        #pragma unroll
        for (int r = 0; r < 8; ++r) {
            int mo = r + 8 * kh;
            Bmid[(rt * 16 + mo) * STRA + colg] = silu_f(acc[r] + bias1);
        }
    }
    __syncthreads();

    // ---- GEMM2: [32,128] @ [128,32]; 4 tiles on waves 0..3 (wave-uniform branch,
    // EXEC stays all-ones inside executing waves as WMMA requires)
    if (wave < 4) {
        const int rt  = wave >> 1;
        const int ct2 = wave & 1;
        const int col2 = ct2 * 16 + m;
        const float bias2 = bb ? bb[col2] : 0.0f;
        v8f acc = {};
        for (int k0 = 0; k0 < HH; k0 += 4) {
            v2f a, b;
            a.x = Bmid[(rt * 16 + m) * STRA + k0 + 2 * kh];
            a.y = Bmid[(rt * 16 + m) * STRA + k0 + 2 * kh + 1];
            b.x = WbL[(k0 + 2 * kh) * STRWB + col2];
            b.y = WbL[(k0 + 2 * kh + 1) * STRWB + col2];
            acc = __builtin_amdgcn_wmma_f32_16x16x4_f32(
                false, a, false, b, (short)0, acc, false, false);
        }
        #pragma unroll
        for (int r = 0; r < 8; ++r) {
            int mo = r + 8 * kh;
            int grow = row0 + rt * 16 + mo;
            if (grow < M) out[(long)grow * DD + col2] = acc[r] + bias2;
        }
    }
}

// ---------------------------------------------------------------------------
// Edge kernel: one wave32 per edge, lane = channel d in [0,32).
// c (51 MB) and h (6.4 MB) live in the 192 MB L2, so the ~1.6 GB of gathers
// never touch HBM; the D=32 l2-norm is a full-wave shfl_xor butterfly.
// ---------------------------------------------------------------------------
__global__ __launch_bounds__(256)
void edge_kernel(const int* __restrict__ ei, const float* __restrict__ rbfs,
                 const float* __restrict__ c, const float* __restrict__ h,
                 float* __restrict__ agg) {
    const int wave = threadIdx.x >> 5;
    const int lane = threadIdx.x & 31;
    const long e = (long)blockIdx.x * 8 + wave;
    if (e >= EE) return;  // wave-uniform

    const int s    = ei[e];        // source (aggregation target)
    const int dvtx = ei[EE + e];   // destination (gather side)

    const float hd = h[(long)dvtx * DD + lane];
    float w = 0.0f;
    #pragma unroll
    for (int r = 0; r < RR; ++r) {
        float cd = c[((long)dvtx * RR + r) * DD + lane];
        float cs = c[((long)s    * RR + r) * DD + lane];
        float ce = cd * cs + cd;
        float sq = ce * ce;
        #pragma unroll
        for (int off = 16; off >= 1; off >>= 1) sq += __shfl_xor(sq, off, 32);
        float nrm = fmaxf(sqrtf(sq), 1e-12f);
        w += rbfs[e * RR + r] * (ce / nrm);
    }
    float sq = w * w;
    #pragma unroll
    for (int off = 16; off >= 1; off >>= 1) sq += __shfl_xor(sq, off, 32);
    float nrm = fmaxf(sqrtf(sq), 1e-12f);
    float msg = hd * (w / nrm);
    atomicAdd(&agg[(long)s * DD + lane], msg);
}

// ---------------------------------------------------------------------------
// Up-projection: out[N,128] = agg[N,32] @ Wu[32,128]; Wu async-staged in LDS.
// ---------------------------------------------------------------------------
__global__ __launch_bounds__(256)
void up_kernel(const float* __restrict__ agg, const float* __restrict__ Wu,
               float* __restrict__ out) {
    __shared__ float At[32 * STRG];
    __shared__ float WuL[DD * STRW];   // 32 x 128 staged weights

    const int tid  = threadIdx.x;
    const int lane = tid & 31;
    const int wave = tid >> 5;
    const int row0 = blockIdx.x * 32;

    // async-stage Wu (32*128 floats = 1024 vec4, 4 issues/thread)
    for (int v4 = tid; v4 < (DD * HH) >> 2; v4 += 256) {
        int r = v4 >> 5;
        int cpos = (v4 & 31) << 2;
        async_copy_f4(&WuL[r * STRW + cpos], Wu + r * HH + cpos);
    }
    // stage agg tile
    for (int idx = tid; idx < 32 * DD; idx += 256) {
        int r = idx >> 5, cc = idx & 31;
        int grow = row0 + r;
        At[r * STRG + cc] = (grow < NN) ? agg[(long)grow * DD + cc] : 0.0f;
    }
    wait_async0();
    __syncthreads();

    const int m  = lane & 15;
    const int kh = lane >> 4;
    const int ct = wave;
    const int colg = ct * 16 + m;
    for (int rt = 0; rt < 2; ++rt) {
        v8f acc = {};
        for (int k0 = 0; k0 < DD; k0 += 4) {
            v2f a, b;
            a.x = At[(rt * 16 + m) * STRG + k0 + 2 * kh];
            a.y = At[(rt * 16 + m) * STRG + k0 + 2 * kh + 1];
            b.x = WuL[(k0 + 2 * kh) * STRW + colg];
            b.y = WuL[(k0 + 2 * kh + 1) * STRW + colg];
            acc = __builtin_amdgcn_wmma_f32_16x16x4_f32(
                false, a, false, b, (short)0, acc, false, false);
        }
        #pragma unroll
        for (int r = 0; r < 8; ++r) {
            int mo = r + 8 * kh;
            int grow = row0 + rt * 16 + mo;
            if (grow < NN) out[(long)grow * HH + colg] = acc[r];
        }
    }
}

// ---------------------------------------------------------------------------
extern "C" void kernel_launch(void* const* d_in, const int* in_sizes, int n_in,
                              void* d_out, int out_size, void* d_ws, size_t ws_size,
                              hipStream_t stream) {
    (void)in_sizes; (void)n_in; (void)out_size; (void)ws_size;

    const float* x      = (const float*)d_in[0];   // [N,H]
    const float* rbfs   = (const float*)d_in[1];   // [E,R]
    const float* coeffs = (const float*)d_in[2];   // [N,R,C] -> flat [N*R, C]
    const float* W1     = (const float*)d_in[3];   // [H,H]
    const float* b1     = (const float*)d_in[4];   // [H]
    const float* W2     = (const float*)d_in[5];   // [H,D]
    const float* b2     = (const float*)d_in[6];   // [D]
    const float* Wc1    = (const float*)d_in[7];   // [C,H]
    const float* Wc2    = (const float*)d_in[8];   // [H,D]
    const float* Wu     = (const float*)d_in[9];   // [D,H]
    const int*   ei     = (const int*)d_in[10];    // [2,E]
    float* out = (float*)d_out;                    // [N,H]

    // Workspace carve-out: h [N,32] | c [N*R,32] | agg [N,32]  (~64 MB)
    float* h   = (float*)d_ws;
    float* c   = h + (size_t)NN * DD;
    float* agg = c + (size_t)NN * RR * DD;

    hipMemsetAsync(agg, 0, (size_t)NN * DD * sizeof(float), stream);

    // node path: h = (silu(silu(x)@W1+b1)) @ W2 + b2
    mlp2_kernel<<<(NN + 31) / 32, 256, 0, stream>>>(x, NN, HH, W1, b1, W2, b2, h);
    // coeffs path: c = silu(silu(coeffs)@Wc1) @ Wc2  (no biases)
    mlp2_kernel<<<(NN * RR + 31) / 32, 256, 0, stream>>>(coeffs, NN * RR, CC,
                                                         Wc1, nullptr, Wc2, nullptr, c);
    // per-edge gather / normalize / scatter-add
    edge_kernel<<<(EE + 7) / 8, 256, 0, stream>>>(ei, rbfs, c, h, agg);
    // up-projection
    up_kernel<<<(NN + 31) / 32, 256, 0, stream>>>(agg, Wu, out);
}